// ChamferLoss_26628797235307
// MI455X (gfx1250) — compile-verified
//
#include <hip/hip_runtime.h>

typedef __attribute__((ext_vector_type(2))) float v2f;
typedef __attribute__((ext_vector_type(8))) float v8f;

// Problem shape from the reference: B=4, N=M=8192, 3-D points.
#define BATCH            4
#define NPTS             8192
#define TILES            (NPTS / 16)          // 512 16-point tiles per batch
#define MTILES           2                    // A-tiles (row tiles) per wave
#define WAVES_PER_BLOCK  8                    // 256 threads = 8 wave32
#define TOTAL_ROWTILES   (BATCH * TILES)      // 2048
#define TOTAL_WAVES      (TOTAL_ROWTILES / MTILES)       // 1024
#define GRID_BLOCKS      (TOTAL_WAVES / WAVES_PER_BLOCK) // 128

// Y is staged in CHUNKS pieces: TDM -> raw (3 floats/pt), then packed once per
// block into B-operand form (-2y0,-2y1,-2y2,||y||^2), 4 floats/pt.
#define CHUNKS       8
#define CHUNK_TILES  (TILES / CHUNKS)        // 64 tiles (even -> pairable)
#define CHUNK_PTS    (CHUNK_TILES * 16)      // 1024 points
#define CHUNK_FLOATS (CHUNK_PTS * 3)         // 3072 floats = 12 KB raw

// ---------- raw min ops (skip fminf's NaN canonicalization) ----------
__device__ __forceinline__ float min_raw(float a, float b) {
    float r;
    asm("v_min_num_f32 %0, %1, %2" : "=v"(r) : "v"(a), "v"(b));
    return r;
}
// 3-input min: folds two distance tiles into the running min in one VOP3 op.
__device__ __forceinline__ float min3_raw(float a, float b, float c) {
    float r;
    asm("v_min3_num_f32 %0, %1, %2, %3" : "=v"(r) : "v"(a), "v"(b), "v"(c));
    return r;
}

// ---------------------- Tensor Data Mover staging --------------------------
#if __has_builtin(__builtin_amdgcn_tensor_load_to_lds)
#define HAVE_TDM 1
typedef unsigned int u32x4 __attribute__((ext_vector_type(4)));
typedef int          i32x4 __attribute__((ext_vector_type(4)));
typedef int          i32x8 __attribute__((ext_vector_type(8)));

// 1-D TDM copy: CHUNK_FLOATS dwords, global -> LDS. Descriptor per
// CDNA5 ISA.
__device__ __forceinline__ void tdm_load_chunk(float* ldsDst, const float* gsrc) {
    const unsigned lds_off = (unsigned)(unsigned long long)ldsDst; // LDS byte offset
    const unsigned long long ga = (unsigned long long)gsrc;
    u32x4 g0 = { 1u,                                   // count=1, user descriptor
                 lds_off,                              // lds_addr
                 (unsigned)(ga & 0xffffffffu),         // global_addr[31:0]
                 (unsigned)((ga >> 32) & 0x01ffffffu)  // global_addr[56:32]
                   | (2u << 30) };                     // type = 2 ("image")
    i32x8 g1 = { (int)(2u << 16),                              // data_size = 4B
                 (int)((CHUNK_FLOATS & 0xffffu) << 16),        // tensor_dim0[15:0]
                 (int)(((CHUNK_FLOATS >> 16) & 0xffffu)        // tensor_dim0[31:16]
                       | (1u << 16)),                          // tensor_dim1 = 1
                 (int)((CHUNK_FLOATS & 0xffffu) << 16),        // tile_dim0
                 1,                                            // tile_dim1 = 1
                 (int)CHUNK_FLOATS,                            // dim0_stride[31:0]
                 0, 0 };
    i32x4 z4 = { 0, 0, 0, 0 };
#if __clang_major__ >= 23
    i32x8 z8 = { 0, 0, 0, 0, 0, 0, 0, 0 };
    __builtin_amdgcn_tensor_load_to_lds(g0, g1, z4, z4, z8, 0);
#else
    __builtin_amdgcn_tensor_load_to_lds(g0, g1, z4, z4, 0);
#endif
}
#else
#define HAVE_TDM 0
#endif

__global__ void chamfer_init(float* accum) { accum[0] = 0.0f; }

// One direction of Chamfer: for each query x in X (B,NPTS,3), min_j ||x-y_j||^2
// over Y (B,NPTS,3); sum of minima atomically accumulated into accum[0].
//
// WMMA mapping (V_WMMA_F32_16X16X4_F32, D = A(16x4)*B(4x16) + C, C = inline 0):
//   A[i,:] = ( x0, x1, x2, 1 ),  B[:,j] = ( -2y0, -2y1, -2y2, ||y_j||^2 )
//   => D[i][j] = ||y_j||^2 - 2<x_i,y_j>   (||x_i||^2 is row-constant, added
//      after the min-reduction). B columns are pre-packed in LDS once per
//      block, so the inner loop is just ds_load_b64 + WMMA + v_min3.
// f32 operand layout: lanes 0-15 hold K=0/K=1 (VGPR0/1), lanes 16-31 K=2/K=3.
__global__ __launch_bounds__(256)
void chamfer_dir(const float* __restrict__ X, const float* __restrict__ Y,
                 float* __restrict__ accum)
{
#if HAVE_TDM
    __shared__ __align__(16) float yraw[2][CHUNK_FLOATS];    // 24 KB raw
#endif
    __shared__ __align__(16) float ypack[2][CHUNK_PTS * 4];  // 32 KB packed
    __shared__ float red[WAVES_PER_BLOCK];

    const int lane = threadIdx.x & 31;
    const int wave = threadIdx.x >> 5;
    const int gw   = blockIdx.x * WAVES_PER_BLOCK + wave;
    const int hl   = lane & 15;
    const bool hi  = lane >= 16;

    const int rowTile0 = gw * MTILES;          // never straddles a batch
    const int b        = rowTile0 / TILES;
    const int rt       = rowTile0 % TILES;

    // ---- A-tiles (query points): embedded (x0,x1,x2,1); keep ||x||^2 aside ----
    v2f   a[MTILES];
    float xnorm[MTILES];
#pragma unroll
    for (int m = 0; m < MTILES; ++m) {
        const float* xp = X + ((size_t)b * NPTS + (size_t)(rt + m) * 16 + hl) * 3;
        const float x0 = xp[0], x1 = xp[1], x2 = xp[2];
        xnorm[m] = x0 * x0 + x1 * x1 + x2 * x2;   // lane L holds row L%16's norm
        a[m].x = hi ? x2   : x0;   // K=2 : K=0
        a[m].y = hi ? 1.0f : x1;   // K=3 : K=1
    }

    const float* ysrc = Y + (size_t)b * NPTS * 3;
    const v8f czero = { 0.f, 0.f, 0.f, 0.f, 0.f, 0.f, 0.f, 0.f }; // -> inline 0

    v8f rm[MTILES];
#pragma unroll
    for (int m = 0; m < MTILES; ++m)
#pragma unroll
        for (int v = 0; v < 8; ++v) rm[m][v] = 3.0e38f;

#if HAVE_TDM
    if (wave == 0) {               // TDM issues once per wave; EXEC ignored
        tdm_load_chunk(&yraw[0][0], ysrc);
        tdm_load_chunk(&yraw[1][0], ysrc + CHUNK_FLOATS);
    }
#endif

    for (int c = 0; c < CHUNKS; ++c) {
        float* __restrict__ pack = &ypack[c & 1][0];
#if HAVE_TDM
        float* __restrict__ raw = &yraw[c & 1][0];
        if (wave == 0) {
            // chunk c complete; at most the next chunk still in flight.
            if (c + 1 < CHUNKS) __builtin_amdgcn_s_wait_tensorcnt(1);
            else                __builtin_amdgcn_s_wait_tensorcnt(0);
        }
        __syncthreads();           // raw[c] visible; pack's old readers done
        // Pack B operand once per block: 4 points per thread per chunk.
        for (int j = threadIdx.x; j < CHUNK_PTS; j += 256) {
            const float t0 = raw[j * 3 + 0], t1 = raw[j * 3 + 1], t2 = raw[j * 3 + 2];
            float4 pk;
            pk.x = -2.0f * t0; pk.y = -2.0f * t1; pk.z = -2.0f * t2;
            pk.w = t0 * t0 + t1 * t1 + t2 * t2;
            *(float4*)&pack[j * 4] = pk;
        }
        __syncthreads();
#else
        __syncthreads();           // pack's old readers done
        for (int j = threadIdx.x; j < CHUNK_PTS; j += 256) {
            const float* gp = ysrc + (size_t)c * CHUNK_FLOATS + (size_t)j * 3;
            const float t0 = gp[0], t1 = gp[1], t2 = gp[2];
            float4 pk;
            pk.x = -2.0f * t0; pk.y = -2.0f * t1; pk.z = -2.0f * t2;
            pk.w = t0 * t0 + t1 * t1 + t2 * t2;
            *(float4*)&pack[j * 4] = pk;
        }
        __syncthreads();
#endif

        // Inner loop: one 8-byte LDS read per lane per tile (lo half: K0,K1;
        // hi half: K2,K3), then WMMA + v_min3. Pairs of tiles share one min3.
        const float* lb = pack + hl * 4 + (hi ? 2 : 0);
#pragma unroll 2
        for (int tt = 0; tt < CHUNK_TILES; tt += 2) {
            const float2 q0 = *(const float2*)(lb + tt * 64);
            const float2 q1 = *(const float2*)(lb + tt * 64 + 64);
            v2f bm0, bm1;
            bm0.x = q0.x; bm0.y = q0.y;
            bm1.x = q1.x; bm1.y = q1.y;
#pragma unroll
            for (int m = 0; m < MTILES; ++m) {
                v8f d0 = __builtin_amdgcn_wmma_f32_16x16x4_f32(
                    false, a[m], false, bm0, (short)0, czero, false, false);
                v8f d1 = __builtin_amdgcn_wmma_f32_16x16x4_f32(
                    false, a[m], false, bm1, (short)0, czero, false, false);
#pragma unroll
                for (int v = 0; v < 8; ++v)
                    rm[m][v] = min3_raw(rm[m][v], d0[v], d1[v]);
            }
        }

        __syncthreads();           // all waves done reading pack (and raw)
#if HAVE_TDM
        if (wave == 0 && (c + 2) < CHUNKS)
            tdm_load_chunk(raw, ysrc + (size_t)(c + 2) * CHUNK_FLOATS);
#endif
    }

    // ---- per-row minima: min across the 16 column-lanes of each half ----
    float s = 0.0f;
#pragma unroll
    for (int m = 0; m < MTILES; ++m) {
#pragma unroll
        for (int v = 0; v < 8; ++v) {
            float r = rm[m][v];
            r = min_raw(r, __shfl_xor(r, 1));
            r = min_raw(r, __shfl_xor(r, 2));
            r = min_raw(r, __shfl_xor(r, 4));
            r = min_raw(r, __shfl_xor(r, 8));
            s += r;   // lanes 0-15: rows 0-7; lanes 16-31: rows 8-15
        }
    }
    s += __shfl_xor(s, 16);        // all lanes: sum of 16 row-minima (both m)

    // Add the deferred ||x||^2 terms: sum over the 16 rows of each A-tile.
#pragma unroll
    for (int m = 0; m < MTILES; ++m) {
        float x = xnorm[m];        // lanes 0-15 hold rows 0-15 (dup in 16-31)
        x += __shfl_xor(x, 1);
        x += __shfl_xor(x, 2);
        x += __shfl_xor(x, 4);
        x += __shfl_xor(x, 8);
        s += x;
    }

    if (lane == 0) red[wave] = s;
    __syncthreads();
    if (threadIdx.x == 0) {
        float tsum = 0.0f;
#pragma unroll
        for (int w = 0; w < WAVES_PER_BLOCK; ++w) tsum += red[w];
        atomicAdd(accum, tsum);
    }
}

__global__ void chamfer_final(const float* __restrict__ accum,
                              float* __restrict__ out)
{
    out[0] = accum[0] * (1.0f / (float)(BATCH * NPTS)); // N == M
}

extern "C" void kernel_launch(void* const* d_in, const int* in_sizes, int n_in,
                              void* d_out, int out_size, void* d_ws, size_t ws_size,
                              hipStream_t stream) {
    (void)in_sizes; (void)n_in; (void)out_size; (void)ws_size;
    const float* pred = (const float*)d_in[0];
    const float* gt   = (const float*)d_in[1];
    float* out   = (float*)d_out;
    float* accum = (float*)d_ws;   // 1 float of scratch

    chamfer_init<<<1, 1, 0, stream>>>(accum);
    chamfer_dir<<<GRID_BLOCKS, 256, 0, stream>>>(pred, gt, accum); // pred -> gt
    chamfer_dir<<<GRID_BLOCKS, 256, 0, stream>>>(gt, pred, accum); // gt -> pred
    chamfer_final<<<1, 1, 0, stream>>>(accum, out);
}